// graphsage_80917183856912
// MI455X (gfx1250) — compile-verified
//
#include <hip/hip_runtime.h>

typedef __attribute__((ext_vector_type(2))) float v2f;
typedef __attribute__((ext_vector_type(8))) float v8f;

#define SA_STRIDE 68    // 16-row A tiles, padded so A-frag ds_load_b64 are bank-conflict-free
#define SW2_UNIT  160   // words per row-pair unit; 160 % 64 == 32 -> halves use disjoint banks

// One layer of GraphSAGE: dst[i,:] = relu( mean_k src[map(nidx[i,k]), :] @ W + bias )
// D = H = 64, K = 32 hardcoded (per reference).
__global__ __launch_bounds__(256) void sage_layer_kernel(
    const float* __restrict__ src,    // [S, 64] source features
    const int*   __restrict__ nidx,   // [rows, 32] neighbor positions
    const int*   __restrict__ remap,  // optional: position -> src row (node ids); may be null
    const float* __restrict__ W,      // [64, 64]
    const float* __restrict__ bias,   // [64]
    float*       __restrict__ dst,    // [rows, 64]
    int rows)
{
    __shared__ float sA[128 * SA_STRIDE];   // up to 8 waves * 16 rows of aggregated features
    __shared__ float sW2[32 * SW2_UNIT];    // W row-pair interleaved: [u][col*2 + (row&1)]

    const int tid    = threadIdx.x;
    const int lane   = tid & 31;
    const int wave   = tid >> 5;
    const int nwaves = blockDim.x >> 5;
    const int l16    = lane & 15;
    const int half   = lane >> 4;

    // Stage W (64x64) into LDS, interleaving adjacent rows so a B fragment
    // (W[brow][c], W[brow+1][c]) is one contiguous, 8B-aligned pair.
    for (int idx = tid; idx < 64 * 64; idx += blockDim.x) {
        const int row = idx >> 6;
        const int col = idx & 63;
        sW2[(row >> 1) * SW2_UNIT + col * 2 + (row & 1)] = W[idx];
    }

    const int rowbase = blockIdx.x * (nwaves * 16) + wave * 16;

    // ---- Phase 1: gather-mean 32 neighbors per row into LDS ----
    for (int j = 0; j < 16; ++j) {
        const int i = rowbase + j;          // wave-uniform
        float ax = 0.f, ay = 0.f;
        if (i < rows) {
            // lane k owns neighbor k of row i
            int nb = nidx[(size_t)i * 32 + lane];
            int sr = remap ? remap[nb] : nb;
            #pragma unroll
            for (int k = 0; k < 32; ++k) {
                int s = __shfl(sr, k, 32);  // broadcast neighbor k's source row
                // each lane loads an 8B slice -> full wave reads the contiguous 256B row
                const v2f v = *(const v2f*)(src + (size_t)s * 64 + lane * 2);
                ax += v.x;
                ay += v.y;
            }
            ax *= (1.f / 32.f);
            ay *= (1.f / 32.f);
        }
        v2f st; st.x = ax; st.y = ay;
        *(v2f*)&sA[(wave * 16 + j) * SA_STRIDE + lane * 2] = st;
    }
    __syncthreads();

    // ---- Phase 2: [16x64] @ [64x64] via V_WMMA_F32_16X16X4_F32 ----
    v8f acc[4];
    {
        v8f z = {0.f, 0.f, 0.f, 0.f, 0.f, 0.f, 0.f, 0.f};
        #pragma unroll
        for (int nt = 0; nt < 4; ++nt) acc[nt] = z;
    }

    const float* aRow = &sA[(wave * 16 + l16) * SA_STRIDE];
    #pragma unroll
    for (int kk = 0; kk < 16; ++kk) {
        // A 16x4 f32 layout: lanes 0-15 hold K=kk*4+{0,1}, lanes 16-31 hold K=kk*4+{2,3}
        const int acol = kk * 4 + half * 2;
        v2f a = *(const v2f*)(aRow + acol);
        // B 4x16: rows (kk*4+2*half, +1) -> one interleaved pair load, no packing movs
        const float* bUnit = &sW2[(kk * 2 + half) * SW2_UNIT];
        #pragma unroll
        for (int nt = 0; nt < 4; ++nt) {
            v2f b = *(const v2f*)(bUnit + (nt * 16 + l16) * 2);
            acc[nt] = __builtin_amdgcn_wmma_f32_16x16x4_f32(
                false, a, false, b, (short)0, acc[nt], false, false);
        }
    }

    // ---- Phase 3: bias + relu + store (C/D layout: VGPR r -> M=r / M=r+8) ----
    #pragma unroll
    for (int nt = 0; nt < 4; ++nt) {
        const int col = nt * 16 + l16;
        const float bv = bias[col];
        #pragma unroll
        for (int r = 0; r < 8; ++r) {
            const int row = rowbase + half * 8 + r;
            if (row < rows) {
                float v = acc[nt][r] + bv;
                dst[(size_t)row * 64 + col] = v > 0.f ? v : 0.f;
            }
        }
    }
}

extern "C" void kernel_launch(void* const* d_in, const int* in_sizes, int n_in,
                              void* d_out, int out_size, void* d_ws, size_t ws_size,
                              hipStream_t stream) {
    // setup_inputs order:
    // 0 feats[V,64] f32, 1 W0[64,64], 2 b0[64], 3 W1[64,64], 4 b1[64],
    // 5 neigh_idx0[N0,32] i32, 6 neigh_idx1[N1,32] i32, 7 node_ids1[N1], 8 node_ids2[N2]
    const float* feats = (const float*)d_in[0];
    const float* W0    = (const float*)d_in[1];
    const float* b0    = (const float*)d_in[2];
    const float* W1    = (const float*)d_in[3];
    const float* b1    = (const float*)d_in[4];
    const int*   nidx0 = (const int*)d_in[5];
    const int*   nidx1 = (const int*)d_in[6];
    const int*   ids2  = (const int*)d_in[8];

    const int N0 = in_sizes[5] / 32;
    const int N1 = in_sizes[6] / 32;

    float* h1  = (float*)d_ws;   // [N1, 64] intermediate (25.6 MB)
    float* out = (float*)d_out;  // [N0, 64]

    // Layer 1: h1 = relu(mean(feats[ids2[nidx1]]) @ W0 + b0)
    // 256 threads = 8 waves = 128 rows/block
    dim3 g1((unsigned)((N1 + 127) / 128));
    sage_layer_kernel<<<g1, dim3(256), 0, stream>>>(feats, nidx1, ids2, W0, b0, h1, N1);

    // Layer 2: out = relu(mean(h1[nidx0]) @ W1 + b1)
    // Smaller blocks (4 waves = 64 rows) for more concurrency on the small N0
    dim3 g2((unsigned)((N0 + 63) / 64));
    sage_layer_kernel<<<g2, dim3(128), 0, stream>>>(h1, nidx0, nullptr, W1, b1, out, N0);
}